// Linear_50302656971673
// MI455X (gfx1250) — compile-verified
//
#include <hip/hip_runtime.h>

// ---------------------------------------------------------------------------
// Shapes
#define B_TOT   4096
#define N_IN    256
#define C_CH    16
#define W_OUT   10
#define WIN     247          // N_IN - W_OUT + 1
#define BN_EPS  1e-5f

// Tiling
#define KC      16           // K-chunk per LDS buffer (double-buffered)
#define NCHUNK  (N_IN / KC)  // 16
#define WPAD    272          // padded filter row: index i -> W[i-15], zero outside
#define BSTRIDE 36           // floats per (b) row in x tile (bank de-alias, 16B-aligned)
#define NLROW   (16 * BSTRIDE)   // 576 floats per j-row
#define BUFSZ   (KC * NLROW)     // 9216 floats per buffer

typedef __attribute__((ext_vector_type(2))) float v2f;
typedef __attribute__((ext_vector_type(8))) float v8f;

#if __has_builtin(__builtin_amdgcn_s_wait_asynccnt)
#define WAIT_ASYNC(n) __builtin_amdgcn_s_wait_asynccnt(n)
#else
#define WAIT_ASYNC(n) asm volatile("s_wait_asynccnt " #n ::: "memory")
#endif

// ---------------------------------------------------------------------------
// Kernel 1: depthwise conv as WMMA GEMM + pointwise chain + raw out + BN partials
__global__ __launch_bounds__(256)
void conv_wmma_kernel(const float* __restrict__ x,      // (B, 256, 16, 2)
                      const float* __restrict__ Wr,     // (16, 247)
                      const float* __restrict__ Wi,     // (16, 247)
                      const float* __restrict__ Wnl,    // (16, 2, 30)
                      const float* __restrict__ Wcoeff, // (2, 2)
                      const float* __restrict__ WoutR,  // (16, 10)
                      const float* __restrict__ WoutI,  // (16, 10)
                      float* __restrict__ outRaw,       // d_out (B, 16, 2)
                      float* __restrict__ P1,           // ws: (256, 16) sum partials
                      float* __restrict__ P2)           // ws: (256, 16) sumsq partials
{
    __shared__ float sW[2 * C_CH * WPAD];   // 8704 floats: zero-padded Bm source
    __shared__ float sX[2 * BUFSZ];         // 18432 floats: double-buffered x tile

    const int tid  = threadIdx.x;
    const int lane = tid & 31;
    const int wv   = tid >> 5;              // wave 0..7 -> channels 2wv, 2wv+1
    const int wg   = blockIdx.x;            // 0..255 batch tiles
    const int b0   = wg * 16;

    const float4* xg = reinterpret_cast<const float4*>(x);

    // ---- issue async prefetch of chunk 0 into buffer 0 --------------------
    // tile element (b, nl, cr4): 16 x KC x 8 float4s, 8 per thread
    {
#pragma unroll
        for (int it = 0; it < 8; ++it) {
            int q    = it * 256 + tid;
            int cr4  = q & 7;
            int rest = q >> 3;
            int nl   = rest & 15;
            int b    = rest >> 4;
            const float4* g = &xg[((size_t)(b0 + b) * N_IN + nl) * 8 + cr4];
            unsigned lds = (unsigned)(unsigned long long)
                           &sX[nl * NLROW + b * BSTRIDE + cr4 * 4];
            asm volatile("global_load_async_to_lds_b128 %0, %1, off"
                         :: "v"(lds), "v"(g) : "memory");
        }
    }

    // ---- stage zero-padded filters (no div/mod): sW[row][i] = W[row][i-15] ----
    for (int row = wv; row < 32; row += 8) {
        const int ri = row >> 4;
        const int c  = row & 15;
        const float* src = (ri ? Wi : Wr) + c * WIN;
        float* dst = &sW[row * WPAD];
        for (int t = lane; t < WPAD; t += 32) {
            int k = t - 15;
            dst[t] = (k >= 0 && k < WIN) ? src[k] : 0.0f;
        }
    }

    v8f acc[4] = {};                        // 2 channels x {real,imag}

    const int half = lane >> 4;             // 0 | 1
    const int bidx = lane & 15;             // A-frag row (batch within tile)
    const int wl   = lane & 15;             // B/D column (output window w)
    const int c0   = wv * 2;

    for (int ch = 0; ch < NCHUNK; ++ch) {
        // ---- async prefetch next chunk into the other buffer ----
        if (ch < NCHUNK - 1) {
            const int n1 = (ch + 1) * KC;
            float* dstB = &sX[((ch + 1) & 1) * BUFSZ];
#pragma unroll
            for (int it = 0; it < 8; ++it) {
                int q    = it * 256 + tid;
                int cr4  = q & 7;
                int rest = q >> 3;
                int nl   = rest & 15;
                int b    = rest >> 4;
                const float4* g = &xg[((size_t)(b0 + b) * N_IN + (n1 + nl)) * 8 + cr4];
                unsigned lds = (unsigned)(unsigned long long)
                               &dstB[nl * NLROW + b * BSTRIDE + cr4 * 4];
                asm volatile("global_load_async_to_lds_b128 %0, %1, off"
                             :: "v"(lds), "v"(g) : "memory");
            }
            WAIT_ASYNC(8);                  // current chunk landed, next in flight
        } else {
            WAIT_ASYNC(0);
        }
        __syncthreads();

        // ---- WMMA K-loop over this chunk ----
        const float* sXc = &sX[(ch & 1) * BUFSZ];
        const int nbase = ch * KC;
#pragma unroll
        for (int kk = 0; kk < KC; kk += 4) {
            const int jl = kk + half * 2;   // per-lane K index within tile
            const int jg = nbase + jl;      // global j
#pragma unroll
            for (int p = 0; p < 4; ++p) {
                const int c  = c0 + (p >> 1);
                const int ri = p & 1;
                const int cr = c * 2 + ri;
                v2f va, vb;
                va.x = sXc[jl * NLROW + bidx * BSTRIDE + cr];
                va.y = sXc[(jl + 1) * NLROW + bidx * BSTRIDE + cr];
                const float* wrow = &sW[(ri * C_CH + c) * WPAD];
                vb.x = wrow[15 + jg - wl];          // Bm[j,   w]
                vb.y = wrow[16 + jg - wl];          // Bm[j+1, w]
                acc[p] = __builtin_amdgcn_wmma_f32_16x16x4_f32(
                    false, va, false, vb, (short)0, acc[p], false, false);
            }
        }
        __syncthreads();                    // buffer may be overwritten next iter
    }

    // ---- amp * f nonlinearity in-register, stage f^3 to LDS (overlay sX) ----
    float* sF  = sX;                        // [2][16][10][16] = 5120 floats
    float* red = sX + 5120;                 // [2][256] reduction scratch
#pragma unroll
    for (int lc = 0; lc < 2; ++lc) {
        const int c = c0 + lc;
        v8f fr  = acc[2 * lc + 0];
        v8f fi  = acc[2 * lc + 1];
        v8f amp = fr * fr + fi * fi;
        v8f f3r = amp * fr;
        v8f f3i = amp * fi;
        if (wl < W_OUT) {
#pragma unroll
            for (int r = 0; r < 8; ++r) {
                int b = r + half * 8;       // C/D layout: M = r + 8*(lane>=16)
                sF[((0 * C_CH + c) * W_OUT + wl) * 16 + b] = f3r[r];
                sF[((1 * C_CH + c) * W_OUT + wl) * 16 + b] = f3i[r];
            }
        }
    }
    __syncthreads();

    // ---- per-(b,c): neighbor mix -> coeff -> output-window reduction ----
    const int b = tid >> 4;
    const int c = tid & 15;
    const float wc00 = Wcoeff[0], wc01 = Wcoeff[1];
    const float wc10 = Wcoeff[2], wc11 = Wcoeff[3];
    float outr = 0.0f, outi = 0.0f;
    for (int w = 0; w < W_OUT; ++w) {
        float no0 = 0.0f, no1 = 0.0f;
#pragma unroll
        for (int j = 0; j < 15; ++j) {
            int cp = (j < c) ? j : j + 1;   // np.delete(arange(16), c)
            float fr = sF[((0 * C_CH + cp) * W_OUT + w) * 16 + b];
            float fi = sF[((1 * C_CH + cp) * W_OUT + w) * 16 + b];
            no0 = fmaf(fr, Wnl[(c * 2 + 0) * 30 + j],      no0);
            no0 = fmaf(fi, Wnl[(c * 2 + 0) * 30 + 15 + j], no0);
            no1 = fmaf(fr, Wnl[(c * 2 + 1) * 30 + j],      no1);
            no1 = fmaf(fi, Wnl[(c * 2 + 1) * 30 + 15 + j], no1);
        }
        float co0 = no0 * wc00 + no1 * wc01;    // p=0
        float co1 = no0 * wc10 + no1 * wc11;    // p=1
        outr = fmaf(co0, WoutR[c * W_OUT + w], outr);
        outi = fmaf(co1, WoutI[c * W_OUT + w], outi);
    }
    const int bg = b0 + b;
    outRaw[(bg * C_CH + c) * 2 + 0] = outr;
    outRaw[(bg * C_CH + c) * 2 + 1] = outi;

    // ---- deterministic per-WG BN partials ----
    red[c * 16 + b]       = outr + outi;
    red[256 + c * 16 + b] = outr * outr + outi * outi;
    __syncthreads();
    if (tid < 16) {
        float s1 = 0.0f, s2 = 0.0f;
        for (int i = 0; i < 16; ++i) {
            s1 += red[tid * 16 + i];
            s2 += red[256 + tid * 16 + i];
        }
        P1[wg * 16 + tid] = s1;
        P2[wg * 16 + tid] = s2;
    }
}

// ---------------------------------------------------------------------------
// Kernel 2: fixed-order reduction of 256 partials per channel -> scale/shift
__global__ void bn_stats_kernel(const float* __restrict__ P1,
                                const float* __restrict__ P2,
                                const float* __restrict__ gamma,
                                const float* __restrict__ beta,
                                float* __restrict__ ss)   // [32]: scale | shift
{
    int c = threadIdx.x;
    if (c < C_CH) {
        float s1 = 0.0f, s2 = 0.0f;
        for (int i = 0; i < 256; ++i) {
            s1 += P1[i * 16 + c];
            s2 += P2[i * 16 + c];
        }
        const float inv = 1.0f / ((float)B_TOT * 2.0f);
        float mu  = s1 * inv;
        float var = s2 * inv - mu * mu;
        float sc  = gamma[c] * rsqrtf(var + BN_EPS);
        ss[c]      = sc;
        ss[16 + c] = beta[c] - mu * sc;
    }
}

// ---------------------------------------------------------------------------
// Kernel 3: in-place normalize of d_out
__global__ __launch_bounds__(256)
void bn_apply_kernel(float* __restrict__ out, const float* __restrict__ ss)
{
    __shared__ float s[32];
    if (threadIdx.x < 32) s[threadIdx.x] = ss[threadIdx.x];
    __syncthreads();
    int idx = blockIdx.x * 512 + threadIdx.x;
#pragma unroll
    for (int i = 0; i < 2; ++i, idx += 256) {
        int c = (idx >> 1) & 15;
        out[idx] = out[idx] * s[c] + s[16 + c];
    }
}

// ---------------------------------------------------------------------------
extern "C" void kernel_launch(void* const* d_in, const int* in_sizes, int n_in,
                              void* d_out, int out_size, void* d_ws, size_t ws_size,
                              hipStream_t stream)
{
    const float* x     = (const float*)d_in[0];
    const float* Wr    = (const float*)d_in[1];
    const float* Wi    = (const float*)d_in[2];
    const float* Wnl   = (const float*)d_in[3];
    const float* Wc    = (const float*)d_in[4];
    const float* WoR   = (const float*)d_in[5];
    const float* WoI   = (const float*)d_in[6];
    const float* gamma = (const float*)d_in[7];
    const float* beta  = (const float*)d_in[8];

    float* out = (float*)d_out;
    float* ws  = (float*)d_ws;
    float* P1  = ws;             // 256*16
    float* P2  = ws + 4096;      // 256*16
    float* SS  = ws + 8192;      // 32

    conv_wmma_kernel<<<256, 256, 0, stream>>>(x, Wr, Wi, Wnl, Wc, WoR, WoI,
                                              out, P1, P2);
    bn_stats_kernel<<<1, 32, 0, stream>>>(P1, P2, gamma, beta, SS);
    bn_apply_kernel<<<256, 256, 0, stream>>>(out, SS);
}